// GATLayer_66941360276235
// MI455X (gfx1250) — compile-verified
//
#include <hip/hip_runtime.h>
#include <stddef.h>

// ---------------------------------------------------------------------------
// GAT attention-logit layer for MI455X (gfx1250, wave32).
//
//   Wh = h @ W ; e1 = Wh @ a1 ; e2 = Wh @ a2 ; out = lrelu(e1[:,None]+e2[None,:])
//
// Restructured as e1 = h @ (W@a1), e2 = h @ (W@a2):
//   K0: b1,b2 = W@a1, W@a2                      (256x64 matvec x2, trivial)
//   K1: [e1 e2] = H @ [b1 b2] via V_WMMA_F32_16X16X4_F32 (fp32 WMMA)
//   K2: out[b,i,j] = max(s, 0.2*s), s = e1[b,i]+e2[b,j]  (256 MB NT stream)
// ---------------------------------------------------------------------------

typedef float v2f __attribute__((ext_vector_type(2)));
typedef float v4f __attribute__((ext_vector_type(4)));
typedef float v8f __attribute__((ext_vector_type(8)));

#define IN_F   256
#define OUT_F  64
#define BATCH  4
#define NPTS   4096
#define NROWS  (BATCH * NPTS)   // 16384 flattened rows of h
#define SLOPE  0.2f

// --- K0: fold a into W:  b1[k] = dot(W[k,:], a[:64]), b2[k] = dot(W[k,:], a[64:])
__global__ void gat_prep(const float* __restrict__ W, const float* __restrict__ a,
                         float* __restrict__ b12) {
    const int k = threadIdx.x;                 // 256 threads, 1 block
    const float* wr = W + k * OUT_F;
    float s1 = 0.0f, s2 = 0.0f;
#pragma unroll
    for (int f = 0; f < OUT_F; ++f) {
        const float w = wr[f];
        s1 += w * a[f];
        s2 += w * a[OUT_F + f];
    }
    b12[k]        = s1;
    b12[IN_F + k] = s2;
}

// --- K1: E(16384x2) = H(16384x256) @ Bv(256x2) using fp32 WMMA 16x16x4.
// One wave per 16-row tile. B-matrix columns 0/1 = b1/b2, columns 2..15 = 0.
__global__ void gat_matvec_wmma(const float* __restrict__ h,
                                const float* __restrict__ b12,
                                float* __restrict__ e12) {
    const int tile = blockIdx.x;               // 1024 tiles
    const int lane = threadIdx.x & 31;
    const int col  = lane & 15;                // A: row-in-tile; B/C/D: column N
    const int khalf = (lane >> 4) << 1;        // 0 (lanes 0-15) or 2 (lanes 16-31)

    const float* __restrict__ hrow = h + (size_t)(tile * 16 + col) * IN_F;
    // lane's B source: col 0 -> b1, col 1 -> b2, cols 2..15 -> masked to 0
    const float* __restrict__ bp  = (col == 0) ? b12 : (b12 + IN_F);
    const float  msk = (col < 2) ? 1.0f : 0.0f;

    v8f c = {0.f, 0.f, 0.f, 0.f, 0.f, 0.f, 0.f, 0.f};

#pragma unroll 4
    for (int kk = 0; kk < IN_F / 4; ++kk) {
        const int kb = kk * 4 + khalf;
        v2f A, B;
        A.x = hrow[kb];
        A.y = hrow[kb + 1];
        B.x = bp[kb]     * msk;
        B.y = bp[kb + 1] * msk;
        // (neg_a, A, neg_b, B, c_mod, C, reuse_a, reuse_b)
        c = __builtin_amdgcn_wmma_f32_16x16x4_f32(false, A, false, B,
                                                  (short)0, c, false, false);
    }

    // D layout: lane L, VGPR r -> M = r + (L>=16 ? 8 : 0), N = L & 15.
    // Column 0 carries e1, column 1 carries e2.
    if (col < 2) {
        float* __restrict__ dst = e12 + (size_t)col * NROWS;
        const int mbase = tile * 16 + ((lane >> 4) << 3);
#pragma unroll
        for (int r = 0; r < 8; ++r) dst[mbase + r] = c[r];
    }
}

// --- K2: the 256 MB streamer. out[b,i,j] = max(s, 0.2*s), s = e1[b,i] + e2[b,j].
// grid = (4, 4096, 4): x = j-chunk (256 thr * float4 = 1024 floats), y = i, z = b.
__global__ void gat_outer(const float* __restrict__ e12, float* __restrict__ out) {
    const int b = blockIdx.z;
    const int i = blockIdx.y;
    const int j = (blockIdx.x * blockDim.x + threadIdx.x) * 4;

    const float* __restrict__ e1 = e12;
    const float* __restrict__ e2 = e12 + NROWS;

    const float s = e1[b * NPTS + i];                          // uniform per block
    const v4f  v  = *(const v4f*)(e2 + b * NPTS + j);          // hot in cache

    v4f r;
    const float x0 = s + v.x, x1 = s + v.y, x2 = s + v.z, x3 = s + v.w;
    r.x = fmaxf(x0, SLOPE * x0);
    r.y = fmaxf(x1, SLOPE * x1);
    r.z = fmaxf(x2, SLOPE * x2);
    r.w = fmaxf(x3, SLOPE * x3);

    const size_t idx = ((size_t)b * NPTS + i) * (size_t)NPTS + j;
    __builtin_nontemporal_store(r, (v4f*)(out + idx));         // NT b128 store
}

extern "C" void kernel_launch(void* const* d_in, const int* in_sizes, int n_in,
                              void* d_out, int out_size, void* d_ws, size_t ws_size,
                              hipStream_t stream) {
    (void)in_sizes; (void)n_in; (void)out_size; (void)ws_size;
    const float* h = (const float*)d_in[0];   // (4, 4096, 256)
    const float* W = (const float*)d_in[1];   // (256, 64)
    const float* a = (const float*)d_in[2];   // (128, 1)
    float* out = (float*)d_out;               // (4, 4096, 4096)

    float* b12 = (float*)d_ws;                // 2 * 256 floats
    float* e12 = b12 + 2 * IN_F;              // 2 * 16384 floats

    gat_prep<<<1, IN_F, 0, stream>>>(W, a, b12);
    gat_matvec_wmma<<<NROWS / 16, 32, 0, stream>>>(h, b12, e12);
    gat_outer<<<dim3(NPTS / 1024, NPTS, BATCH), 256, 0, stream>>>(e12, out);
}